// GCN_53927609368950
// MI455X (gfx1250) — compile-verified
//
#include <hip/hip_runtime.h>
#include <hip/hip_bf16.h>
#include <math.h>

typedef __attribute__((ext_vector_type(2))) float v2f;
typedef __attribute__((ext_vector_type(8))) float v8f;

static __host__ int cdiv(int a, int b) { return (a + b - 1) / b; }

// ---------------- degree / dinv ----------------
__global__ void gcn_zero_u32(unsigned* p, int n) {
    int i = blockIdx.x * blockDim.x + threadIdx.x;
    if (i < n) p[i] = 0u;
}

__global__ void gcn_degree(const int* __restrict__ ei, unsigned* __restrict__ deg, int e) {
    int i = blockIdx.x * blockDim.x + threadIdx.x;
    if (i < e) atomicAdd(&deg[ei[(size_t)e + i]], 1u);
}

__global__ void gcn_dinv(const unsigned* __restrict__ deg, float* __restrict__ dinv, int n) {
    int i = blockIdx.x * blockDim.x + threadIdx.x;
    if (i < n) dinv[i] = rsqrtf((float)(deg[i] + 1u));   // +1 self loop; always > 0
}

// ---------------- WMMA GEMM: C[n x DOUT] = A[n x 128] @ W[128 x DOUT] ----------------
// 16 rows per block, one wave per 16-wide column tile. fp32 WMMA 16x16x4, K=128.
template <int DOUT>
__global__ __launch_bounds__(DOUT * 2)
void gcn_gemm_wmma(const float* __restrict__ A, const float* __restrict__ W,
                   float* __restrict__ C, int n) {
    constexpr int NW = DOUT / 16;       // waves per block
    constexpr int NT = NW * 32;         // threads per block
    __shared__ float Alds[16][132];     // pitch 132: bank-conflict-free column reads

    const int rowBase = blockIdx.x * 16;
    const int tid = threadIdx.x;

    // Stage A tile (16 x 128 fp32) into LDS with float4 loads.
    for (int i = tid; i < 16 * 32; i += NT) {
        int r = i >> 5, c4 = i & 31;
        float4 v = make_float4(0.f, 0.f, 0.f, 0.f);
        if (rowBase + r < n)
            v = ((const float4*)(A + (size_t)(rowBase + r) * 128))[c4];
        Alds[r][c4 * 4 + 0] = v.x; Alds[r][c4 * 4 + 1] = v.y;
        Alds[r][c4 * 4 + 2] = v.z; Alds[r][c4 * 4 + 3] = v.w;
    }
    __syncthreads();

    const int wave = tid >> 5;
    const int lane = tid & 31;
    const int colBase = wave * 16;
    const int hk = (lane >> 4) << 1;    // 0 (lanes 0-15) or 2 (lanes 16-31)
    const int r16 = lane & 15;

    v8f acc = {};
#pragma unroll
    for (int k0 = 0; k0 < 128; k0 += 4) {
        v2f a, b;
        // A 16x4 fragment: lane<16 -> A[r16][k0..k0+1]; lane>=16 -> A[r16][k0+2..k0+3]
        a.x = Alds[r16][k0 + hk];
        a.y = Alds[r16][k0 + hk + 1];
        // B 4x16 fragment (row striped across lanes): rows k0+hk, k0+hk+1
        b.x = W[(size_t)(k0 + hk) * DOUT + colBase + r16];
        b.y = W[(size_t)(k0 + hk + 1) * DOUT + colBase + r16];
        acc = __builtin_amdgcn_wmma_f32_16x16x4_f32(
            /*neg_a=*/false, a, /*neg_b=*/false, b,
            /*c_mod=*/(short)0, acc, /*reuse_a=*/false, /*reuse_b=*/false);
    }

    // D layout: VGPR r holds M=r (lanes 0-15) and M=r+8 (lanes 16-31), N=lane&15
    const int rowOff = (lane >> 4) * 8;
    const int col = colBase + r16;
#pragma unroll
    for (int r = 0; r < 8; ++r) {
        int row = rowBase + r + rowOff;
        if (row < n) C[(size_t)row * DOUT + col] = acc[r];
    }
}

// ---------------- aggregation: agg = self-loop init, then edge scatter ----------------
template <int D>   // D = 128 or 64
__global__ void gcn_init_agg(const float* __restrict__ hw, const float* __restrict__ dinv,
                             const float* __restrict__ bias, float* __restrict__ agg, int n) {
    constexpr int C4 = D / 4;           // float4 per row
    int i = blockIdx.x * blockDim.x + threadIdx.x;
    if (i >= n * C4) return;
    int row = i / C4, c4 = i % C4;
    float d = dinv[row];
    float s = d * d;
    float4 v = ((const float4*)hw)[i];
    float4 b = ((const float4*)bias)[c4];
    float4 o = make_float4(fmaf(v.x, s, b.x), fmaf(v.y, s, b.y),
                           fmaf(v.z, s, b.z), fmaf(v.w, s, b.w));
    ((float4*)agg)[i] = o;
}

template <int D>   // one wave per edge; lane covers D/32 contiguous floats
__global__ void gcn_edge_agg(const int* __restrict__ ei, const float* __restrict__ dinv,
                             const float* __restrict__ hw, float* __restrict__ agg, int e) {
    int edge = blockIdx.x * (blockDim.x >> 5) + (threadIdx.x >> 5);
    if (edge >= e) return;
    int lane = threadIdx.x & 31;
    int src = ei[edge];
    int dst = ei[(size_t)e + edge];
    float nrm = dinv[src] * dinv[dst];
    if constexpr (D == 128) {
        float4 v = ((const float4*)(hw + (size_t)src * D))[lane];
        float* a = agg + (size_t)dst * D + lane * 4;
        atomicAdd(&a[0], v.x * nrm);
        atomicAdd(&a[1], v.y * nrm);
        atomicAdd(&a[2], v.z * nrm);
        atomicAdd(&a[3], v.w * nrm);
    } else {
        float2 v = ((const float2*)(hw + (size_t)src * D))[lane];
        float* a = agg + (size_t)dst * D + lane * 2;
        atomicAdd(&a[0], v.x * nrm);
        atomicAdd(&a[1], v.y * nrm);
    }
}

// ---------------- BatchNorm ----------------
__global__ void gcn_zero_f32(float* p, int n) {
    int i = blockIdx.x * blockDim.x + threadIdx.x;
    if (i < n) p[i] = 0.f;
}

__global__ void gcn_bn_stats(const float* __restrict__ h, float* __restrict__ stats,
                             int n, int rowsPerBlock) {
    int f = threadIdx.x;                    // 128 features
    int r0 = blockIdx.x * rowsPerBlock;
    int r1 = min(r0 + rowsPerBlock, n);
    float s = 0.f, s2 = 0.f;
    for (int r = r0; r < r1; ++r) {
        float v = h[(size_t)r * 128 + f];
        s += v; s2 += v * v;
    }
    atomicAdd(&stats[f], s);
    atomicAdd(&stats[128 + f], s2);
}

__global__ void gcn_bn_finalize(float* __restrict__ stats, const float* __restrict__ g,
                                const float* __restrict__ be, float invN) {
    int f = threadIdx.x;                    // 128
    float m = stats[f] * invN;
    float var = fmaxf(stats[128 + f] * invN - m * m, 0.f);
    float rstd = rsqrtf(var + 1e-5f);
    float sc = rstd * g[f];
    stats[256 + f] = sc;                    // scale
    stats[384 + f] = be[f] - m * sc;        // shift
}

__global__ void gcn_bn_apply_relu(float* __restrict__ h, const float* __restrict__ sc,
                                  const float* __restrict__ sh, int n) {
    int i = blockIdx.x * blockDim.x + threadIdx.x;   // over n*32 float4 groups
    if (i >= n * 32) return;
    int f4 = i & 31;
    float4 v = ((float4*)h)[i];
    float4 s = ((const float4*)sc)[f4];
    float4 t = ((const float4*)sh)[f4];
    v.x = fmaxf(fmaf(v.x, s.x, t.x), 0.f);
    v.y = fmaxf(fmaf(v.y, s.y, t.y), 0.f);
    v.z = fmaxf(fmaf(v.z, s.z, t.z), 0.f);
    v.w = fmaxf(fmaf(v.w, s.w, t.w), 0.f);
    ((float4*)h)[i] = v;
}

// ---------------- softmax over 64 cols, one wave per row ----------------
__global__ void gcn_softmax64(const float* __restrict__ h, float* __restrict__ out, int n) {
    int row = blockIdx.x * (blockDim.x >> 5) + (threadIdx.x >> 5);
    if (row >= n) return;
    int lane = threadIdx.x & 31;
    float2 v = ((const float2*)(h + (size_t)row * 64))[lane];
    float m = fmaxf(v.x, v.y);
#pragma unroll
    for (int o = 16; o > 0; o >>= 1) m = fmaxf(m, __shfl_xor(m, o, 32));
    float e0 = __expf(v.x - m), e1 = __expf(v.y - m);
    float s = e0 + e1;
#pragma unroll
    for (int o = 16; o > 0; o >>= 1) s += __shfl_xor(s, o, 32);
    float inv = 1.f / s;
    ((float2*)(out + (size_t)row * 64))[lane] = make_float2(e0 * inv, e1 * inv);
}

// ---------------- driver ----------------
extern "C" void kernel_launch(void* const* d_in, const int* in_sizes, int n_in,
                              void* d_out, int out_size, void* d_ws, size_t ws_size,
                              hipStream_t stream) {
    const float* x   = (const float*)d_in[0];
    const int*   ei  = (const int*)d_in[1];
    const float* W1  = (const float*)d_in[2];
    const float* b1  = (const float*)d_in[3];
    const float* W2  = (const float*)d_in[4];
    const float* b2  = (const float*)d_in[5];
    const float* W3  = (const float*)d_in[6];
    const float* b3  = (const float*)d_in[7];
    const float* g1  = (const float*)d_in[8];
    const float* be1 = (const float*)d_in[9];
    const float* g2  = (const float*)d_in[10];
    const float* be2 = (const float*)d_in[11];
    float* out = (float*)d_out;

    const int n = in_sizes[0] / 128;   // 100000
    const int e = in_sizes[1] / 2;     // 3200000

    // workspace layout
    float*    hw    = (float*)d_ws;                    // n*128
    float*    agg   = hw  + (size_t)n * 128;           // n*128
    float*    dinv  = agg + (size_t)n * 128;           // n
    float*    stats = dinv + n;                        // 512: sum|sumsq|scale|shift
    unsigned* deg   = (unsigned*)(stats + 512);        // n

    const int RPB = 256;                               // rows per BN-stats block
    const float invN = 1.f / (float)n;

    // degree -> dinv
    gcn_zero_u32<<<cdiv(n, 256), 256, 0, stream>>>(deg, n);
    gcn_degree<<<cdiv(e, 256), 256, 0, stream>>>(ei, deg, e);
    gcn_dinv<<<cdiv(n, 256), 256, 0, stream>>>(deg, dinv, n);

    // ---- layer 1 ----
    gcn_gemm_wmma<128><<<cdiv(n, 16), 256, 0, stream>>>(x, W1, hw, n);
    gcn_init_agg<128><<<cdiv(n * 32, 256), 256, 0, stream>>>(hw, dinv, b1, agg, n);
    gcn_edge_agg<128><<<cdiv(e, 8), 256, 0, stream>>>(ei, dinv, hw, agg, e);
    gcn_zero_f32<<<2, 256, 0, stream>>>(stats, 512);
    gcn_bn_stats<<<cdiv(n, RPB), 128, 0, stream>>>(agg, stats, n, RPB);
    gcn_bn_finalize<<<1, 128, 0, stream>>>(stats, g1, be1, invN);
    gcn_bn_apply_relu<<<cdiv(n * 32, 256), 256, 0, stream>>>(agg, stats + 256, stats + 384, n);

    // ---- layer 2 ----
    gcn_gemm_wmma<128><<<cdiv(n, 16), 256, 0, stream>>>(agg, W2, hw, n);
    gcn_init_agg<128><<<cdiv(n * 32, 256), 256, 0, stream>>>(hw, dinv, b2, agg, n);
    gcn_edge_agg<128><<<cdiv(e, 8), 256, 0, stream>>>(ei, dinv, hw, agg, e);
    gcn_zero_f32<<<2, 256, 0, stream>>>(stats, 512);
    gcn_bn_stats<<<cdiv(n, RPB), 128, 0, stream>>>(agg, stats, n, RPB);
    gcn_bn_finalize<<<1, 128, 0, stream>>>(stats, g2, be2, invN);
    gcn_bn_apply_relu<<<cdiv(n * 32, 256), 256, 0, stream>>>(agg, stats + 256, stats + 384, n);

    // ---- layer 3 (DOUT=64) + softmax ----
    gcn_gemm_wmma<64><<<cdiv(n, 16), 128, 0, stream>>>(agg, W3, hw, n);
    gcn_init_agg<64><<<cdiv(n * 16, 256), 256, 0, stream>>>(hw, dinv, b3, agg, n);
    gcn_edge_agg<64><<<cdiv(e, 8), 256, 0, stream>>>(ei, dinv, hw, agg, e);
    gcn_softmax64<<<cdiv(n, 8), 256, 0, stream>>>(agg, out, n);
}